// CPDTail_27135603376206
// MI455X (gfx1250) — compile-verified
//
#include <hip/hip_runtime.h>

// CPDTail: depthwise CP-decomposed conv along height.
// x: (B=16, Hp=58, C=128, R=8, W=56) f32
// weight: (C=128, R=8, K=3) f32, bias: (C,) f32
// out: (B, C, h=56, W=56) f32
//
// out[b,c,h,w] = sum_{kt<3} sum_{r<8} x[b,h+kt,c,r,w] * weight[c,r,kt] + bias[c]
//
// Memory-bound (~1.3 FLOP/byte, ~239 MB min traffic, ~10us at 23.3 TB/s).
// Rank/tap contraction is done with V_WMMA_F32_16X16X4_F32:
//   A(16x4) = weight chunk broadcast across all 16 rows
//   B(4x16) = x for 4 rank rows x 16 w positions
//   D(16x16): all rows identical; row 0 (VGPR0, lanes 0-15) = 16 outputs.

typedef float v2f __attribute__((ext_vector_type(2)));
typedef float v8f __attribute__((ext_vector_type(8)));

#define B_DIM 16
#define HP    58
#define C_DIM 128
#define R_DIM 8
#define W_DIM 56
#define HOUT  56

__global__ __launch_bounds__(256) void cpd_tail_wmma_kernel(
    const float* __restrict__ x, const float* __restrict__ weight,
    const float* __restrict__ bias, float* __restrict__ out)
{
  const int bc   = blockIdx.x;        // bc = b*C + c
  const int b    = bc >> 7;           // / 128
  const int c    = bc & 127;
  const int lane = (int)(threadIdx.x & 31u);
  const int wave = (int)(threadIdx.x >> 5);   // 0..7
  const int half = (lane < 16) ? 0 : 2;       // K-half this lane supplies
  const int wlo  = lane & 15;

  // Preload the 6 A-fragments (kt in 0..2, rh in 0..1 -> r0 = 4*rh).
  // A[m][k] = weight[c, r0+k, kt] broadcast over m; lane half selects K pair.
  float a0[6], a1[6];
#pragma unroll
  for (int kt = 0; kt < 3; ++kt) {
#pragma unroll
    for (int rh = 0; rh < 2; ++rh) {
      const int q = kt * 2 + rh;
      const int r = rh * 4 + half;
      a0[q] = weight[c * (R_DIM * 3) + r * 3 + kt];
      a1[q] = weight[c * (R_DIM * 3) + (r + 1) * 3 + kt];
    }
  }
  const float bias_c = bias[c];

  float* __restrict__ outp = out + (size_t)bc * (HOUT * W_DIM);

  // 56 h-rows x 4 w-tiles of 16 = 224 tiles, round-robin over 8 waves.
  for (int t = wave; t < HOUT * 4; t += 8) {
    const int h  = t >> 2;
    const int wb = (t & 3) << 4;
    int wcol = wb + wlo;
    if (wcol > W_DIM - 1) wcol = W_DIM - 1;   // clamp: clamped columns never stored

    v8f acc = {bias_c, bias_c, bias_c, bias_c, bias_c, bias_c, bias_c, bias_c};

#pragma unroll
    for (int kt = 0; kt < 3; ++kt) {
      const size_t rowbase =
          ((size_t)(b * HP + h + kt) * C_DIM + c) * (size_t)(R_DIM * W_DIM);
#pragma unroll
      for (int rh = 0; rh < 2; ++rh) {
        const int q = kt * 2 + rh;
        const size_t base = rowbase + (size_t)(rh * 4 + half) * W_DIM + wcol;
        v2f Bv;
        Bv.x = x[base];            // B[K=half][n]
        Bv.y = x[base + W_DIM];    // B[K=half+1][n]
        v2f Av;
        Av.x = a0[q];
        Av.y = a1[q];
        // D = A(16x4) * B(4x16) + C  -- f32 WMMA, 8-arg form:
        // (neg_a, A, neg_b, B, c_mod, C, reuse_a, reuse_b)
        acc = __builtin_amdgcn_wmma_f32_16x16x4_f32(
            false, Av, false, Bv, (short)0, acc, false, false);
      }
    }

    // Row M=0 of D lives in acc[0] of lanes 0..15 (N = lane).
    if (lane < 16 && (wb + lane) < W_DIM) {
      outp[h * W_DIM + wb + lane] = acc[0];
    }
  }
}

extern "C" void kernel_launch(void* const* d_in, const int* in_sizes, int n_in,
                              void* d_out, int out_size, void* d_ws, size_t ws_size,
                              hipStream_t stream) {
  (void)in_sizes; (void)n_in; (void)out_size; (void)d_ws; (void)ws_size;
  const float* x      = (const float*)d_in[0];
  const float* weight = (const float*)d_in[1];
  const float* bias   = (const float*)d_in[2];
  float* out          = (float*)d_out;

  dim3 grid(B_DIM * C_DIM);   // one block per (b, c)
  dim3 block(256);            // 8 waves
  hipLaunchKernelGGL(cpd_tail_wmma_kernel, grid, block, 0, stream,
                     x, weight, bias, out);
}